// EEGConvNetMiniV3_7112465842810
// MI455X (gfx1250) — compile-verified
//
#include <hip/hip_runtime.h>

typedef __attribute__((ext_vector_type(2))) float v2f;
typedef __attribute__((ext_vector_type(8))) float v8f;

#define LRELU_SLOPE 0.01f
#define BN_EPS 1e-5f

__device__ __forceinline__ float lrelu(float v){ return v > 0.f ? v : LRELU_SLOPE * v; }
__device__ __forceinline__ unsigned f2key(float f){
  unsigned u = __float_as_uint(f);
  return (u & 0x80000000u) ? ~u : (u | 0x80000000u);   // monotonic: bigger float -> bigger key
}

// ---------------- GEMM 1: [N x 128] @ [128 x 16] via V_WMMA_F32_16X16X4_F32 ----------------
__global__ void gemm_x_w1(const float* __restrict__ x, const float* __restrict__ W,
                          float* __restrict__ out, int nTiles){
  int wave = (blockIdx.x * blockDim.x + threadIdx.x) >> 5;
  if (wave >= nTiles) return;                 // wave-uniform: EXEC stays all-ones
  int lane = threadIdx.x & 31;
  int m = lane & 15, kh = lane >> 4;
  long r0 = (long)wave * 16;
  const float* xr = x + (r0 + m) * 128;
  v8f c = {};
  #pragma unroll
  for (int k0 = 0; k0 < 128; k0 += 4){
    int ka = k0 + 2*kh;
    v2f a, b;
    a.x = xr[ka];            a.y = xr[ka + 1];
    b.x = W[ka*16 + m];      b.y = W[(ka + 1)*16 + m];
    c = __builtin_amdgcn_wmma_f32_16x16x4_f32(false, a, false, b, (short)0, c, false, false);
  }
  #pragma unroll
  for (int j = 0; j < 8; ++j)
    out[(r0 + kh*8 + j)*16 + m] = c[j];
}

// ---------------- GEMM 2: [K1 x 16] @ [16 x 32] via V_WMMA_F32_16X16X4_F32 ----------------
__global__ void gemm_xk_w2(const float* __restrict__ xk, const float* __restrict__ W,
                           float* __restrict__ out, int nRowTiles){
  int wave = (blockIdx.x * blockDim.x + threadIdx.x) >> 5;
  if (wave >= nRowTiles * 2) return;
  int lane = threadIdx.x & 31;
  int m = lane & 15, kh = lane >> 4;
  int rt = wave >> 1, c0 = (wave & 1) * 16;
  long r0 = (long)rt * 16;
  const float* ar = xk + (r0 + m) * 16;
  v8f c = {};
  #pragma unroll
  for (int k0 = 0; k0 < 16; k0 += 4){
    int ka = k0 + 2*kh;
    v2f a, b;
    a.x = ar[ka];                 a.y = ar[ka + 1];
    b.x = W[ka*32 + c0 + m];      b.y = W[(ka + 1)*32 + c0 + m];
    c = __builtin_amdgcn_wmma_f32_16x16x4_f32(false, a, false, b, (short)0, c, false, false);
  }
  #pragma unroll
  for (int j = 0; j < 8; ++j)
    out[(r0 + kh*8 + j)*32 + c0 + m] = c[j];
}

// ---------------- edge scatter-add, 16 channels (full graph) ----------------
__global__ void scatter16(const int* __restrict__ src, const int* __restrict__ dst,
                          const float* __restrict__ h, float* __restrict__ agg, int E){
  int e = blockIdx.x * blockDim.x + threadIdx.x;
  if (e >= E) return;
  const float4* hs = (const float4*)(h + (long)src[e] * 16);
  float* ad = agg + (long)dst[e] * 16;
  #pragma unroll
  for (int q = 0; q < 4; ++q){
    float4 v = hs[q];
    unsafeAtomicAdd(ad + q*4 + 0, v.x);
    unsafeAtomicAdd(ad + q*4 + 1, v.y);
    unsafeAtomicAdd(ad + q*4 + 2, v.z);
    unsafeAtomicAdd(ad + q*4 + 3, v.w);
  }
}

// ---------------- edge scatter-add, 32 channels through inv remap (pooled graph) ----------------
__global__ void scatter32_inv(const int* __restrict__ src, const int* __restrict__ dst,
                              const int* __restrict__ inv, const float* __restrict__ h,
                              float* __restrict__ agg, int E){
  int e = blockIdx.x * blockDim.x + threadIdx.x;
  if (e >= E) return;
  int s = inv[src[e]], d = inv[dst[e]];
  if (s < 0 || d < 0) return;                 // emask==0 for dropped edges
  const float4* hs = (const float4*)(h + (long)s * 32);
  float* ad = agg + (long)d * 32;
  #pragma unroll
  for (int q = 0; q < 8; ++q){
    float4 v = hs[q];
    unsafeAtomicAdd(ad + q*4 + 0, v.x);
    unsafeAtomicAdd(ad + q*4 + 1, v.y);
    unsafeAtomicAdd(ad + q*4 + 2, v.z);
    unsafeAtomicAdd(ad + q*4 + 3, v.w);
  }
}

// ---------------- BatchNorm: per-channel sum / sumsq (LDS atomics + global flush) ----------------
__global__ void bn_stats(const float* __restrict__ h, long total, int C, float* __restrict__ stats){
  __shared__ float s[64];
  for (int i = threadIdx.x; i < 2*C; i += blockDim.x) s[i] = 0.f;
  __syncthreads();
  for (long i = (long)blockIdx.x * blockDim.x + threadIdx.x; i < total;
       i += (long)gridDim.x * blockDim.x){
    float v = h[i];
    int c = (int)(i & (long)(C - 1));
    atomicAdd(&s[c], v);
    atomicAdd(&s[C + c], v * v);
  }
  __syncthreads();
  for (int i = threadIdx.x; i < 2*C; i += blockDim.x) unsafeAtomicAdd(&stats[i], s[i]);
}

__global__ void bn_apply(float* __restrict__ h, long total, int C, float invN,
                         const float* __restrict__ stats, const float* __restrict__ g,
                         const float* __restrict__ be){
  long i = (long)blockIdx.x * blockDim.x + threadIdx.x;
  if (i >= total) return;
  int c = (int)(i & (long)(C - 1));
  float mu  = stats[c] * invN;
  float var = stats[C + c] * invN - mu * mu;   // biased variance
  float v = (h[i] - mu) * rsqrtf(var + BN_EPS) * g[c] + be[c];
  h[i] = lrelu(v);
}

// ---------------- SAG score: agg@Wr + br + h@Wroot, plus sortable key ----------------
template <int C>
__global__ void score_kernel(const float* __restrict__ agg, const float* __restrict__ h,
                             const float* __restrict__ Wr, const float* __restrict__ br,
                             const float* __restrict__ Wroot, float* __restrict__ score,
                             unsigned* __restrict__ keys, int n){
  int i = blockIdx.x * blockDim.x + threadIdx.x;
  if (i >= n) return;
  float acc = br[0];
  const float* a = agg + (long)i * C;
  const float* xr = h + (long)i * C;
  #pragma unroll
  for (int c = 0; c < C; ++c) acc += a[c] * Wr[c] + xr[c] * Wroot[c];
  score[i] = acc;
  keys[i]  = f2key(acc);
}

// ---------------- exact top-k: 4-pass radix select on 8-bit digits ----------------
__global__ void rs_init(unsigned* state, int k){ state[0] = 0u; state[1] = (unsigned)k; }

__global__ void rs_hist(const unsigned* __restrict__ keys, int n, const unsigned* __restrict__ state,
                        unsigned* __restrict__ hist, int pass){
  unsigned prefix = state[0];
  int done = 8 * pass;
  for (int i = blockIdx.x * blockDim.x + threadIdx.x; i < n; i += gridDim.x * blockDim.x){
    unsigned key = keys[i];
    bool match = (pass == 0) || ((key >> (32 - done)) == (prefix >> (32 - done)));
    if (match) atomicAdd(&hist[(key >> (24 - done)) & 255u], 1u);
  }
}

__global__ void rs_select(unsigned* state, const unsigned* __restrict__ hist, int pass){
  unsigned kneed = state[1], cum = 0;
  int chosen = 0;
  for (int b = 255; b >= 0; --b){
    unsigned c = hist[b];
    if (cum + c >= kneed){ chosen = b; break; }
    cum += c;
  }
  state[0] |= ((unsigned)chosen) << (24 - 8 * pass);
  state[1]  = kneed - cum;
}

__global__ void rs_compact(const unsigned* __restrict__ keys, int n, const unsigned* __restrict__ state,
                           int k, int* __restrict__ perm, unsigned* __restrict__ counters){
  unsigned T = state[0], tneed = state[1];
  unsigned G = (unsigned)k - tneed;           // count of keys strictly greater than threshold
  for (int i = blockIdx.x * blockDim.x + threadIdx.x; i < n; i += gridDim.x * blockDim.x){
    unsigned key = keys[i];
    if (key > T){
      unsigned p = atomicAdd(&counters[0], 1u);
      perm[p] = i;
    } else if (key == T){
      unsigned t = atomicAdd(&counters[1], 1u);
      if (t < tneed) perm[G + t] = i;
    }
  }
}

// ---------------- pool-1 gather: xk = h[perm]*tanh(score[perm]); inv[perm[j]] = j ----------------
__global__ void build_pool16(const int* __restrict__ perm, const float* __restrict__ h,
                             const float* __restrict__ score, float* __restrict__ xk,
                             int* __restrict__ inv, int k){
  int j = blockIdx.x * blockDim.x + threadIdx.x;
  if (j >= k) return;
  int node = perm[j];
  inv[node] = j;
  float t = tanhf(score[node]);
  const float4* hr = (const float4*)(h + (long)node * 16);
  float4* xo = (float4*)(xk + (long)j * 16);
  #pragma unroll
  for (int q = 0; q < 4; ++q){
    float4 v = hr[q];
    v.x *= t; v.y *= t; v.z *= t; v.w *= t;
    xo[q] = v;
  }
}

// ---------------- global add pool over top-k2 (gated), 32 channels ----------------
__global__ void pool_sum32(const int* __restrict__ perm, const float* __restrict__ h,
                           const float* __restrict__ score, float* __restrict__ pooled, int k){
  __shared__ float s[32];
  if (threadIdx.x < 32) s[threadIdx.x] = 0.f;
  __syncthreads();
  for (int j = blockIdx.x * blockDim.x + threadIdx.x; j < k; j += gridDim.x * blockDim.x){
    int node = perm[j];
    float t = tanhf(score[node]);
    const float* hr = h + (long)node * 32;
    #pragma unroll
    for (int c = 0; c < 32; ++c) atomicAdd(&s[c], hr[c] * t);
  }
  __syncthreads();
  if (threadIdx.x < 32) unsafeAtomicAdd(&pooled[threadIdx.x], s[threadIdx.x]);
}

// ---------------- tiny MLP head: 32 -> 8 -> 4 -> 2, leaky relu ----------------
__global__ void mlp_head(const float* __restrict__ pooled,
                         const float* __restrict__ fw1, const float* __restrict__ fb1,
                         const float* __restrict__ fw2, const float* __restrict__ fb2,
                         const float* __restrict__ fw3, const float* __restrict__ fb3,
                         float* __restrict__ out){
  if (threadIdx.x != 0 || blockIdx.x != 0) return;
  float a1[8], a2[4];
  for (int o = 0; o < 8; ++o){
    float acc = fb1[o];
    for (int i = 0; i < 32; ++i) acc += pooled[i] * fw1[i*8 + o];
    a1[o] = lrelu(acc);
  }
  for (int o = 0; o < 4; ++o){
    float acc = fb2[o];
    for (int i = 0; i < 8; ++i) acc += a1[i] * fw2[i*4 + o];
    a2[o] = lrelu(acc);
  }
  for (int o = 0; o < 2; ++o){
    float acc = fb3[o];
    for (int i = 0; i < 4; ++i) acc += a2[i] * fw3[i*2 + o];
    out[o] = lrelu(acc);
  }
}

extern "C" void kernel_launch(void* const* d_in, const int* in_sizes, int n_in,
                              void* d_out, int out_size, void* d_ws, size_t ws_size,
                              hipStream_t stream){
  (void)in_sizes; (void)n_in; (void)out_size; (void)ws_size;
  const int N = 200000, E = 6400000;
  const int K1 = 100000, K2 = 50000;        // ceil(N/2), ceil(K1/2)

  const float* x   = (const float*)d_in[0];
  const int*   ei  = (const int*)d_in[1];
  const int*   src = ei;
  const int*   dst = ei + E;
  // d_in[2] edge_weigth unused: reference uses emask = ones
  // d_in[3] batch unused (single-graph semantics)
  const float* W1     = (const float*)d_in[4];
  // b1 (d_in[5]) cancels exactly inside training-mode BN
  const float* g1     = (const float*)d_in[6];
  const float* be1    = (const float*)d_in[7];
  const float* Wr1    = (const float*)d_in[8];
  const float* br1    = (const float*)d_in[9];
  const float* Wroot1 = (const float*)d_in[10];
  const float* W2     = (const float*)d_in[11];
  // b2 (d_in[12]) cancels inside BN
  const float* g2     = (const float*)d_in[13];
  const float* be2    = (const float*)d_in[14];
  const float* Wr2    = (const float*)d_in[15];
  const float* br2    = (const float*)d_in[16];
  const float* Wroot2 = (const float*)d_in[17];
  const float* fw1 = (const float*)d_in[18]; const float* fb1 = (const float*)d_in[19];
  const float* fw2 = (const float*)d_in[20]; const float* fb2 = (const float*)d_in[21];
  const float* fw3 = (const float*)d_in[22]; const float* fb3 = (const float*)d_in[23];

  char* ws = (char*)d_ws;
  size_t off = 0;
  auto alloc = [&](size_t bytes) -> char* {
    off = (off + 255) & ~(size_t)255;
    char* p = ws + off;
    off += bytes;
    return p;
  };
  float*    bufA    = (float*)alloc((size_t)N  * 16 * 4);  // h1lin, then score-agg
  float*    bufB    = (float*)alloc((size_t)N  * 16 * 4);  // agg1 -> h1 (in place)
  float*    score1  = (float*)alloc((size_t)N  * 4);
  unsigned* keys1   = (unsigned*)alloc((size_t)N * 4);
  int*      inv1    = (int*)alloc((size_t)N * 4);
  int*      perm1   = (int*)alloc((size_t)K1 * 4);
  float*    xk      = (float*)alloc((size_t)K1 * 16 * 4);
  float*    h2b     = (float*)alloc((size_t)K1 * 32 * 4);  // h2lin, then score2-agg
  float*    h2c     = (float*)alloc((size_t)K1 * 32 * 4);  // agg2 -> h2 (in place)
  float*    score2  = (float*)alloc((size_t)K1 * 4);
  unsigned* keys2   = (unsigned*)alloc((size_t)K1 * 4);
  int*      perm2   = (int*)alloc((size_t)K2 * 4);
  float*    stats   = (float*)alloc(64 * 4);
  unsigned* hist    = (unsigned*)alloc(256 * 4);
  unsigned* state   = (unsigned*)alloc(4 * 4);
  unsigned* counters= (unsigned*)alloc(2 * 4);
  float*    pooled  = (float*)alloc(32 * 4);

  const int B = 256;
  auto gs = [&](long n){ return (int)((n + B - 1) / B); };

  // --- conv1: h1lin = x @ W1 (WMMA f32) ---
  { int tiles = N / 16; gemm_x_w1<<<gs((long)tiles * 32), B, 0, stream>>>(x, W1, bufA, tiles); }
  // --- agg1 = scatter(h1lin) ---
  hipMemsetAsync(bufB, 0, (size_t)N * 16 * 4, stream);
  scatter16<<<gs(E), B, 0, stream>>>(src, dst, bufA, bufB, E);
  // --- BN1 + leaky relu (bias cancels in BN) ---
  hipMemsetAsync(stats, 0, 64 * 4, stream);
  bn_stats<<<1024, B, 0, stream>>>(bufB, (long)N * 16, 16, stats);
  bn_apply<<<gs((long)N * 16), B, 0, stream>>>(bufB, (long)N * 16, 16, 1.f / N, stats, g1, be1);
  // --- SAG score 1 ---
  hipMemsetAsync(bufA, 0, (size_t)N * 16 * 4, stream);
  scatter16<<<gs(E), B, 0, stream>>>(src, dst, bufB, bufA, E);
  score_kernel<16><<<gs(N), B, 0, stream>>>(bufA, bufB, Wr1, br1, Wroot1, score1, keys1, N);
  // --- exact top-K1 radix select + compact ---
  rs_init<<<1, 1, 0, stream>>>(state, K1);
  for (int p = 0; p < 4; ++p){
    hipMemsetAsync(hist, 0, 256 * 4, stream);
    rs_hist<<<512, B, 0, stream>>>(keys1, N, state, hist, p);
    rs_select<<<1, 1, 0, stream>>>(state, hist, p);
  }
  hipMemsetAsync(counters, 0, 2 * 4, stream);
  rs_compact<<<512, B, 0, stream>>>(keys1, N, state, K1, perm1, counters);
  // --- pool 1: xk = h1[perm]*tanh(score), inv map ---
  hipMemsetAsync(inv1, 0xFF, (size_t)N * 4, stream);
  build_pool16<<<gs(K1), B, 0, stream>>>(perm1, bufB, score1, xk, inv1, K1);
  // --- conv2: h2lin = xk @ W2 (WMMA f32) ---
  { int rt = K1 / 16; gemm_xk_w2<<<gs((long)rt * 2 * 32), B, 0, stream>>>(xk, W2, h2b, rt); }
  // --- agg2 over remapped edges ---
  hipMemsetAsync(h2c, 0, (size_t)K1 * 32 * 4, stream);
  scatter32_inv<<<gs(E), B, 0, stream>>>(src, dst, inv1, h2b, h2c, E);
  // --- BN2 + leaky relu ---
  hipMemsetAsync(stats, 0, 64 * 4, stream);
  bn_stats<<<1024, B, 0, stream>>>(h2c, (long)K1 * 32, 32, stats);
  bn_apply<<<gs((long)K1 * 32), B, 0, stream>>>(h2c, (long)K1 * 32, 32, 1.f / K1, stats, g2, be2);
  // --- SAG score 2 ---
  hipMemsetAsync(h2b, 0, (size_t)K1 * 32 * 4, stream);
  scatter32_inv<<<gs(E), B, 0, stream>>>(src, dst, inv1, h2c, h2b, E);
  score_kernel<32><<<gs(K1), B, 0, stream>>>(h2b, h2c, Wr2, br2, Wroot2, score2, keys2, K1);
  // --- exact top-K2 ---
  rs_init<<<1, 1, 0, stream>>>(state, K2);
  for (int p = 0; p < 4; ++p){
    hipMemsetAsync(hist, 0, 256 * 4, stream);
    rs_hist<<<512, B, 0, stream>>>(keys2, K1, state, hist, p);
    rs_select<<<1, 1, 0, stream>>>(state, hist, p);
  }
  hipMemsetAsync(counters, 0, 2 * 4, stream);
  rs_compact<<<512, B, 0, stream>>>(keys2, K1, state, K2, perm2, counters);
  // --- global add pool (gated) + MLP head ---
  hipMemsetAsync(pooled, 0, 32 * 4, stream);
  pool_sum32<<<256, B, 0, stream>>>(perm2, h2c, score2, pooled, K2);
  mlp_head<<<1, 1, 0, stream>>>(pooled, fw1, fb1, fw2, fb2, fw3, fb3, (float*)d_out);
}